// DWNModel_26731876450941
// MI455X (gfx1250) — compile-verified
//
#include <hip/hip_runtime.h>
#include <hip/hip_bf16.h>

typedef __attribute__((ext_vector_type(2))) float v2f;
typedef __attribute__((ext_vector_type(8))) float v8f;

#define B_SZ   512
#define F_SZ   1024
#define T_SZ   3
#define I_SZ   (F_SZ * T_SZ)      // 3072 bits -> 96 u32 words
#define NW     (I_SZ / 32)        // 96
#define O1_SZ  2000
#define O2_SZ  1000
#define NCLS   10
#define GRP    (O2_SZ / NCLS)     // 100  (divisible by 4 -> per-chunk class is uniform)

// ---------------------------------------------------------------------------
// Kernel A: per-batch-row fused pipeline.
//   stage 1: thermometer-encode x[b,:] vs thresholds -> 96 packed u32 in LDS
//   stage 2: layer-1 exact LUT lookup (bits are exactly 0/1) -> h1[2000] in LDS
//   stage 3: layer-2 6-D multilinear interpolation (factorized 8x8) -> h2 global
// ---------------------------------------------------------------------------
__global__ void __launch_bounds__(256) dwn_fused_kernel(
    const float* __restrict__ x,      // (512,1024)
    const float* __restrict__ th,     // (1024,3)
    const float* __restrict__ luts1,  // (2000,64)
    const int*   __restrict__ idx1,   // (2000,6)
    const float* __restrict__ luts2,  // (1000,64)
    const int*   __restrict__ idx2,   // (1000,6)
    float* __restrict__ h2)           // (512,1000) workspace
{
    __shared__ unsigned int bits[NW];
    __shared__ float h1s[O1_SZ];

    const int b   = blockIdx.x;
    const int tid = threadIdx.x;
    const float* xr = x + (size_t)b * F_SZ;

    // ---- stage 1: pack thermometer bits (bit j = f*3 + t) -----------------
    if (tid < NW) {
        unsigned int w = 0;
        int j = tid * 32;
        #pragma unroll 4
        for (int i = 0; i < 32; ++i, ++j) {
            int f = j / 3;
            int t = j - f * 3;
            w |= (xr[f] > th[f * 3 + t] ? 1u : 0u) << i;
        }
        bits[tid] = w;
    }
    __syncthreads();

    // ---- stage 2: layer 1 (exact lookup; k=0 is the MSB of the index) -----
    for (int o = tid; o < O1_SZ; o += 256) {
        const int* ip = idx1 + o * 6;
        unsigned int index = 0;
        #pragma unroll
        for (int k = 0; k < 6; ++k) {
            int j = ip[k];
            unsigned int bit = (bits[j >> 5] >> (j & 31)) & 1u;
            index |= bit << (5 - k);
        }
        h1s[o] = luts1[o * 64 + index];
    }
    __syncthreads();

    // ---- stage 3: layer 2 multilinear interp, factorized wH[8] x wL[8] ----
    for (int o = tid; o < O2_SZ; o += 256) {
        const int* ip = idx2 + o * 6;
        float xk[6];
        #pragma unroll
        for (int k = 0; k < 6; ++k) xk[k] = h1s[ip[k]];

        float wH[8], wL[8];
        #pragma unroll
        for (int i = 0; i < 8; ++i) {
            float p = ((i & 4) ? xk[0] : (1.0f - xk[0]))
                    * ((i & 2) ? xk[1] : (1.0f - xk[1]))
                    * ((i & 1) ? xk[2] : (1.0f - xk[2]));
            float q = ((i & 4) ? xk[3] : (1.0f - xk[3]))
                    * ((i & 2) ? xk[4] : (1.0f - xk[4]))
                    * ((i & 1) ? xk[5] : (1.0f - xk[5]));
            wH[i] = p;
            wL[i] = q;
        }

        const float4* lp = (const float4*)(luts2 + o * 64);  // 256B aligned
        float acc = 0.0f;
        #pragma unroll
        for (int i = 0; i < 8; ++i) {
            float4 lo = lp[i * 2];
            float4 hi = lp[i * 2 + 1];
            float s = lo.x * wL[0] + lo.y * wL[1] + lo.z * wL[2] + lo.w * wL[3]
                    + hi.x * wL[4] + hi.y * wL[5] + hi.z * wL[6] + hi.w * wL[7];
            acc += wH[i] * s;
        }
        h2[(size_t)b * O2_SZ + o] = acc;
    }
}

// ---------------------------------------------------------------------------
// Kernel B: group_sum as a GEMM:  out(512x10) = h2(512x1000) x S(1000x16)/tau
// One wave32 per 16 batch rows; 250 x V_WMMA_F32_16X16X4_F32 over K=1000.
// Class boundaries (multiples of 100) never split a K=4 chunk, so the selector
// B matrix is hoisted: outer loop over the 10 classes (one compare each),
// inner loop of 25 chunks is pure  global_load_b64 + v_wmma.
// A 16x4 f32 layout: lanes 0-15 -> M=lane, K rows {0,1}; lanes 16-31 -> K {2,3}.
// B  4x16 f32 layout: lanes 0-15 -> N=lane, K rows {0,1}; lanes 16-31 -> K {2,3}.
// C/D: VGPR r holds M=r (lanes 0-15) / M=r+8 (lanes 16-31), N = lane&15.
// ---------------------------------------------------------------------------
__global__ void __launch_bounds__(32) groupsum_wmma_kernel(
    const float* __restrict__ h2,     // (512,1000)
    float* __restrict__ out)          // (512,10)
{
    const int lane = threadIdx.x;                 // 0..31, EXEC all ones
    const int b0   = blockIdx.x * 16;
    const int m    = lane & 15;                   // M row (A) / N col (B)
    const int koff = (lane >> 4) * 2;             // 0 (lanes 0-15) or 2 (16-31)

    // float2-granular A pointer: element kbase/2 == rows {kbase, kbase+1}
    const v2f* rowp =
        (const v2f*)(h2 + (size_t)(b0 + m) * O2_SZ + koff);

    v8f c = {};
    for (int cls = 0; cls < NCLS; ++cls) {
        const float sel = (cls == m) ? 1.0f : 0.0f;
        v2f bv;  bv.x = sel;  bv.y = sel;         // uniform within each K=4 chunk
        const v2f* ap = rowp + cls * (GRP / 2);   // start of this class's 100 K rows
        #pragma unroll 5
        for (int kc = 0; kc < GRP / 4; ++kc) {    // 25 chunks of K=4
            v2f a = ap[kc * 2];                   // this lane's 2 K rows of the chunk
            // D = A x B + C  (8 args: neg_a, A, neg_b, B, c_mod, C, reuse_a, reuse_b)
            c = __builtin_amdgcn_wmma_f32_16x16x4_f32(
                    false, a, false, bv, (short)0, c, false, false);
        }
    }

    const float invtau = 1.0f / 3.3333333f;
    #pragma unroll
    for (int r = 0; r < 8; ++r) {
        int M = r + ((lane >> 4) << 3);           // lanes 0-15: M=r ; 16-31: M=r+8
        if (m < NCLS)
            out[(size_t)(b0 + M) * NCLS + m] = c[r] * invtau;
    }
}

// ---------------------------------------------------------------------------
extern "C" void kernel_launch(void* const* d_in, const int* in_sizes, int n_in,
                              void* d_out, int out_size, void* d_ws, size_t ws_size,
                              hipStream_t stream) {
    const float* x     = (const float*)d_in[0];
    const float* th    = (const float*)d_in[1];
    const float* luts1 = (const float*)d_in[2];
    const int*   idx1  = (const int*)  d_in[3];
    const float* luts2 = (const float*)d_in[4];
    const int*   idx2  = (const int*)  d_in[5];
    float* h2  = (float*)d_ws;                    // 512*1000*4 = 2 MB scratch
    float* out = (float*)d_out;                   // 512*10

    dwn_fused_kernel<<<B_SZ, 256, 0, stream>>>(x, th, luts1, idx1, luts2, idx2, h2);
    groupsum_wmma_kernel<<<B_SZ / 16, 32, 0, stream>>>(h2, out);
}